// CompositionalPhoneticsModel_9182640078998
// MI455X (gfx1250) — compile-verified
//
#include <hip/hip_runtime.h>
#include <math.h>

typedef __attribute__((ext_vector_type(2))) float v2f;
typedef __attribute__((ext_vector_type(8))) float v8f;

#define B_    8
#define T_    1024
#define H_    640
#define NP    230           // phones
#define NPP   240           // phones padded to 16
#define NM    96            // phonemes
#define ROWS  (B_ * T_)     // 8192
#define MT    (ROWS / 16)   // 512 M tiles
#define NT    (NPP / 16)    // 15 N tiles
#define SCALE 0.0395284707521047416f   // 1/sqrt(640)

// -------- Stage 0: pad + transpose + pre-scale B.
// Bt[p][k] = f2p[k][p] / sqrt(H) for p < 230, else 0.  (240 x 640)
__global__ __launch_bounds__(256) void pad_b(
    const float* __restrict__ f2p,    // (640, 230)
    float* __restrict__ bt)           // (240, 640)
{
  int tid = blockIdx.x * 256 + threadIdx.x;   // 0 .. 240*640-1
  int p = tid / H_;
  int k = tid - p * H_;
  float v = (p < NP) ? f2p[(size_t)k * NP + p] * SCALE : 0.0f;
  bt[(size_t)p * H_ + k] = v;
}

// -------- Stage 1: fp32 WMMA GEMM  phone[r][p] = enc[r][:] . Bt[p][:]
// One wave per 16x16 tile; inner loop = 2x global_load_b64 + 1x wmma.
__global__ __launch_bounds__(256) void phone_gemm_wmma(
    const float* __restrict__ enc,      // (8192, 640)
    const float* __restrict__ bt,       // (240, 640) transposed, padded, scaled
    float* __restrict__ phone)          // (8192, 240) padded
{
  const int wave  = blockIdx.x * 8 + (threadIdx.x >> 5);
  const int lane  = threadIdx.x & 31;
  const int half  = lane >> 4;     // 0: K pair {0,1}, 1: K pair {2,3}
  const int l16   = lane & 15;
  const int mTile = wave & (MT - 1);
  const int nTile = wave >> 9;     // 0..14

  const int rowA = mTile * 16 + l16;     // A: lane supplies row M=l16
  const int colB = nTile * 16 + l16;     // B: lane supplies col N=l16 (< 240, in-bounds)

  const float* arow = enc + (size_t)rowA * H_ + 2 * half;
  const float* brow = bt  + (size_t)colB * H_ + 2 * half;

  v8f acc = {};
  #pragma unroll 8
  for (int k = 0; k < H_; k += 4) {
    // A fragment (16x4 f32): v0=K(k+2h), v1=K(k+2h+1)  -> one b64 load
    v2f a = *(const v2f*)(arow + k);
    // B fragment (4x16 f32): same K striping, contiguous in Bt -> one b64 load
    v2f b = *(const v2f*)(brow + k);
    acc = __builtin_amdgcn_wmma_f32_16x16x4_f32(
        /*neg_a=*/false, a, /*neg_b=*/false, b,
        /*c_mod=*/(short)0, acc, /*reuse_a=*/false, /*reuse_b=*/false);
  }

  // C/D layout: VGPR i -> (M = i + 8*half, N = l16)
  const int col = nTile * 16 + l16;
  #pragma unroll
  for (int i = 0; i < 8; ++i) {
    const int r = mTile * 16 + i + 8 * half;
    phone[(size_t)r * NPP + col] = acc[i];
  }
}

// -------- Stage 2: masked segment-max over phones + log-softmax over phonemes.
// 96 threads per block (3 waves), thread m = phoneme m.
__global__ __launch_bounds__(96) void phoneme_reduce(
    const float* __restrict__ phone,     // (8192, 240)
    const float* __restrict__ mapping,   // (96, 230) 0/1 floats
    float* __restrict__ out)             // (8192, 96)
{
  __shared__ float sphone[NPP];
  __shared__ float sred[NM];
  const int m = threadIdx.x;             // 0..95

  // Fold this phoneme's mapping row into a 230-bit register mask (once per block).
  unsigned mask[8];
  #pragma unroll
  for (int i = 0; i < 8; ++i) mask[i] = 0u;
  for (int p = 0; p < NP; ++p)
    if (mapping[(size_t)m * NP + p] > 0.0f) mask[p >> 5] |= (1u << (p & 31));

  for (int row = blockIdx.x; row < ROWS; row += gridDim.x) {
    for (int p = threadIdx.x; p < NPP; p += NM)
      sphone[p] = phone[(size_t)row * NPP + p];
    __syncthreads();

    // Masked max over phones for this phoneme (LDS broadcast reads).
    float best = -INFINITY;
    for (int p = 0; p < NP; ++p)
      if ((mask[p >> 5] >> (p & 31)) & 1u) best = fmaxf(best, sphone[p]);

    // log-softmax across the 96 phonemes (cross-wave via LDS).
    sred[m] = best;
    __syncthreads();
    float gmax = -INFINITY;
    #pragma unroll 4
    for (int i = 0; i < NM; ++i) gmax = fmaxf(gmax, sred[i]);
    __syncthreads();
    sred[m] = expf(best - gmax);
    __syncthreads();
    float gsum = 0.0f;
    #pragma unroll 4
    for (int i = 0; i < NM; ++i) gsum += sred[i];
    __syncthreads();

    out[(size_t)row * NM + m] = best - gmax - logf(gsum);
  }
}

extern "C" void kernel_launch(void* const* d_in, const int* in_sizes, int n_in,
                              void* d_out, int out_size, void* d_ws, size_t ws_size,
                              hipStream_t stream) {
  const float* enc = (const float*)d_in[0];   // (8,1024,640) f32
  const float* f2p = (const float*)d_in[1];   // (640,230)    f32
  const float* map = (const float*)d_in[2];   // (96,230)     f32
  float* out   = (float*)d_out;               // (8,1024,96)  f32

  float* phone = (float*)d_ws;                         // (8192,240) f32 = 7.5 MiB
  float* bt    = phone + (size_t)ROWS * NPP;           // (240,640)  f32 = 600 KiB

  pad_b<<<dim3((NPP * H_) / 256), dim3(256), 0, stream>>>(f2p, bt);
  phone_gemm_wmma<<<dim3((MT * NT) / 8), dim3(256), 0, stream>>>(enc, bt, phone);
  phoneme_reduce<<<dim3(1024), dim3(96), 0, stream>>>(phone, map, out);
}